// LP_PN_61237643706598
// MI455X (gfx1250) — compile-verified
//
#include <hip/hip_runtime.h>

// ---------------- constants ----------------
#define NTOT   4000
#define NS     500
#define NQ     3500
#define DIMD   300
#define WAYS   20
#define SHOTS  25
#define GK     20
#define LDP    32            // padded RHS columns (20 live)
#define ALPHA_C 0.99f
#define EPS_C   2.2204460492503131e-16f
#define BNEPS   1e-5f
#define CG_ITERS 128

typedef float v2f __attribute__((ext_vector_type(2)));
typedef float v8f __attribute__((ext_vector_type(8)));

// ---------------- workspace offsets (bytes) ----------------
static const size_t OFF_MAT    = 0;            // 64,000,000  W -> S ; reused for attention h1a/h2a after CG
static const size_t OFF_MASK   = 64000000;     // 16,000,000  byte mask ; reused for out3 (5.6MB)
static const size_t OFF_E      = 80000000;     // 4,800,000
static const size_t OFF_H1     = 84800000;     // 4,800,000   gw conv1 out
static const size_t OFF_H2     = 89600000;     // 64,000      gw conv2 out
static const size_t OFF_SIG    = 89664000;     // 16,000
static const size_t OFF_N2     = 89680000;     // 16,000
static const size_t OFF_BN     = 89696000;     // 1,024   [0..74]=scale1 [75..149]=shift1 [150]=scale2 [151]=shift2
static const size_t OFF_TOPG   = 89697024;     // 320,000 int
static const size_t OFF_DSI    = 90017024;     // 16,000
static const size_t OFF_R      = 90033024;     // 512,000
static const size_t OFF_P      = 90545024;     // 512,000
static const size_t OFF_V      = 91057024;     // 512,000
static const size_t OFF_AP     = 91569024;     // 512,000
static const size_t OFF_X      = 92081024;     // 512,000  (Fmat)
static const size_t OFF_CGS    = 92593024;     // 1,024  floats: rr@0 pap@32 alpha@64 beta@96 cs@128
static const size_t OFF_QIDX   = 92594048;     // 2,048 int
static const size_t OFF_CENTER = 92596096;     // 24,000
static const size_t OFF_PROTO  = 92620096;     // 24,000
static const size_t OFF_HAL    = 92644096;     // 96,000
static const size_t OFF_LAM    = 92740096;     // 256
static const size_t OFF_AUP    = 92740352;     // 24,000
static const size_t OFF_SCORE  = 92764352;     // 24,000
static const size_t OFF_PRED   = 92788352;     // 280,000

// ---------------- helpers ----------------
__device__ __forceinline__ float embv(const float* XS, const float* XQ, int n, int d) {
    return (n < NS) ? XS[n * DIMD + d] : XQ[(n - NS) * DIMD + d];
}

__device__ __forceinline__ float bred(float v) {
    __shared__ float sh[256];
    int t = threadIdx.x;
    sh[t] = v; __syncthreads();
    for (int s = blockDim.x >> 1; s > 0; s >>= 1) {
        if (t < s) sh[t] += sh[t + s];
        __syncthreads();
    }
    float r = sh[0]; __syncthreads();
    return r;
}

// ============================================================
// graph_weight  (maxpools are identity for 2x2 input, pad 1)
// ============================================================
__global__ void k_gw_conv1(const float* XS, const float* XQ, const float* w, const float* b, float* h1) {
    int idx = blockIdx.x * blockDim.x + threadIdx.x;
    if (idx >= NTOT * 75 * 4) return;
    int pos = idx & 3, o = (idx >> 2) % 75, n = idx / 300;
    int i = pos >> 1, j = pos & 1;
    float acc = b[o];
    for (int c = 0; c < 75; ++c)
        for (int di = 0; di < 3; ++di) {
            int ii = i + di - 1; if (ii < 0 || ii > 1) continue;
            for (int dj = 0; dj < 3; ++dj) {
                int jj = j + dj - 1; if (jj < 0 || jj > 1) continue;
                acc += embv(XS, XQ, n, c * 4 + ii * 2 + jj) * w[((o * 75 + c) * 3 + di) * 3 + dj];
            }
        }
    h1[idx] = acc;
}

__global__ void k_gw_bn1(const float* h1, const float* g, const float* be, float* bn) {
    int o = blockIdx.x;                       // 75 blocks
    float s = 0.f, ss = 0.f;
    for (int t = threadIdx.x; t < NTOT * 4; t += blockDim.x) {
        float v = h1[(t >> 2) * 300 + o * 4 + (t & 3)];
        s += v; ss += v * v;
    }
    s = bred(s); ss = bred(ss);
    if (threadIdx.x == 0) {
        float m = s / (NTOT * 4.f), var = ss / (NTOT * 4.f) - m * m;
        float sc = g[o] * rsqrtf(var + BNEPS);
        bn[o] = sc; bn[75 + o] = be[o] - m * sc;
    }
}

__global__ void k_gw_conv2(const float* h1, const float* bn, const float* w2, const float* b2, float* h2) {
    int idx = blockIdx.x * blockDim.x + threadIdx.x;
    if (idx >= NTOT * 4) return;
    int pos = idx & 3, n = idx >> 2;
    int i = pos >> 1, j = pos & 1;
    float acc = b2[0];
    for (int c = 0; c < 75; ++c) {
        float sc = bn[c], sh = bn[75 + c];
        for (int di = 0; di < 3; ++di) {
            int ii = i + di - 1; if (ii < 0 || ii > 1) continue;
            for (int dj = 0; dj < 3; ++dj) {
                int jj = j + dj - 1; if (jj < 0 || jj > 1) continue;
                float v = fmaxf(h1[n * 300 + c * 4 + ii * 2 + jj] * sc + sh, 0.f);
                acc += v * w2[(c * 3 + di) * 3 + dj];
            }
        }
    }
    h2[idx] = acc;
}

__global__ void k_gw_bn2(const float* h2, const float* g, const float* be, float* bn) {
    float s = 0.f, ss = 0.f;
    for (int t = threadIdx.x; t < NTOT * 4; t += blockDim.x) {
        float v = h2[t]; s += v; ss += v * v;
    }
    s = bred(s); ss = bred(ss);
    if (threadIdx.x == 0) {
        float m = s / (NTOT * 4.f), var = ss / (NTOT * 4.f) - m * m;
        float sc = g[0] * rsqrtf(var + BNEPS);
        bn[150] = sc; bn[151] = be[0] - m * sc;
    }
}

__global__ void k_gw_head(const float* h2, const float* bn, const float* w3, const float* b3,
                          const float* w4, const float* b4, float* sig) {
    int n = blockIdx.x * blockDim.x + threadIdx.x;
    if (n >= NTOT) return;
    float v[4];
    for (int k = 0; k < 4; ++k) v[k] = fmaxf(h2[n * 4 + k] * bn[150] + bn[151], 0.f);
    float s = b4[0];
    for (int j = 0; j < 16; ++j) {
        float h = b3[j];
        for (int k = 0; k < 4; ++k) h += v[k] * w3[j * 4 + k];
        s += fmaxf(h, 0.f) * w4[j];
    }
    sig[n] = s;
}

__global__ void k_E(const float* XS, const float* XQ, const float* sig, float* E, float* n2) {
    int n = blockIdx.x;
    float inv = 1.f / (sig[n] + EPS_C);
    float loc = 0.f;
    for (int d = threadIdx.x; d < DIMD; d += blockDim.x) {
        float e = embv(XS, XQ, n, d) * inv;
        E[n * DIMD + d] = e; loc += e * e;
    }
    loc = bred(loc);
    if (threadIdx.x == 0) n2[n] = loc;
}

// ============================================================
// Gram:  W = exp(-max(n2i+n2j-2*E.Et,0)/(2D))   (f32 WMMA)
// ============================================================
__global__ void k_gram(const float* __restrict__ E, const float* __restrict__ n2, float* __restrict__ W) {
    int lane = threadIdx.x & 31, wave = threadIdx.x >> 5;
    int row0 = (blockIdx.y * 2 + (wave >> 1)) * 16;
    int col0 = (blockIdx.x * 2 + (wave & 1)) * 16;
    int m = lane & 15;
    int kb = (lane >> 4) * 2;
    const float* arow = E + (size_t)(row0 + m) * DIMD;
    const float* brow = E + (size_t)(col0 + m) * DIMD;   // B[k][n] = E[col0+n][k]; n == lane&15 == m
    __builtin_prefetch(arow, 0, 3);
    __builtin_prefetch(brow, 0, 3);
    v8f acc = {0.f, 0.f, 0.f, 0.f, 0.f, 0.f, 0.f, 0.f};
    #pragma unroll 5
    for (int k = 0; k < DIMD; k += 4) {
        v2f a = *reinterpret_cast<const v2f*>(arow + k + kb);
        v2f b = *reinterpret_cast<const v2f*>(brow + k + kb);
        acc = __builtin_amdgcn_wmma_f32_16x16x4_f32(false, a, false, b, (short)0, acc, false, false);
    }
    int hi = (lane >> 4) * 8, n = lane & 15;
    #pragma unroll
    for (int v = 0; v < 8; ++v) {
        int i = row0 + v + hi, j = col0 + n;
        float sq = fmaxf(n2[i] + n2[j] - 2.f * acc[v], 0.f) * (1.f / DIMD);
        W[(size_t)i * NTOT + j] = expf(-0.5f * sq);
    }
}

// ---------------- top-G per row ----------------
__global__ void k_topG(const float* W, int* topg) {
    __shared__ float vals[NTOT];
    __shared__ float sv[256];
    __shared__ int   si[256];
    int row = blockIdx.x, t = threadIdx.x;
    for (int j = t; j < NTOT; j += blockDim.x) vals[j] = W[(size_t)row * NTOT + j];
    __syncthreads();
    for (int g = 0; g < GK; ++g) {
        float bv = -1e30f; int bi = 0;
        for (int j = t; j < NTOT; j += blockDim.x) {
            float v = vals[j];
            if (v > bv || (v == bv && j < bi)) { bv = v; bi = j; }
        }
        sv[t] = bv; si[t] = bi; __syncthreads();
        for (int s = blockDim.x >> 1; s > 0; s >>= 1) {
            if (t < s) {
                if (sv[t + s] > sv[t] || (sv[t + s] == sv[t] && si[t + s] < si[t])) {
                    sv[t] = sv[t + s]; si[t] = si[t + s];
                }
            }
            __syncthreads();
        }
        if (t == 0) { topg[row * GK + g] = si[0]; vals[si[0]] = -1e30f; }
        __syncthreads();
    }
}

__global__ void k_mask_clear(unsigned int* m32) {
    int i = blockIdx.x * blockDim.x + threadIdx.x;
    if (i < NTOT * NTOT / 4) m32[i] = 0u;
}
__global__ void k_mask_scatter(const int* topg, unsigned char* mask) {
    int t = blockIdx.x * blockDim.x + threadIdx.x;
    if (t >= NTOT * GK) return;
    int i = t / GK, j = topg[t];
    mask[(size_t)i * NTOT + j] = 1;
    mask[(size_t)j * NTOT + i] = 1;
}
__global__ void k_wmask(float* W, const unsigned char* mask) {
    size_t i = (size_t)blockIdx.x * blockDim.x + threadIdx.x;
    if (i < (size_t)NTOT * NTOT && !mask[i]) W[i] = 0.f;
}
__global__ void k_dv(const float* W, float* dsi) {   // W symmetric -> col sums == row sums
    int i = blockIdx.x;
    float s = 0.f;
    for (int j = threadIdx.x; j < NTOT; j += blockDim.x) s += W[(size_t)i * NTOT + j];
    s = bred(s);
    if (threadIdx.x == 0) dsi[i] = sqrtf(1.f / (s + EPS_C));
}
__global__ void k_scaleS(float* W, const float* dsi) {
    size_t t = (size_t)blockIdx.x * blockDim.x + threadIdx.x;
    if (t >= (size_t)NTOT * NTOT) return;
    int i = (int)(t / NTOT), j = (int)(t % NTOT);
    W[t] *= dsi[i] * dsi[j];
}

// ============================================================
// CG solve of (I - alpha*S + eps*11^T) X = Y  (20 RHS, padded 32)
// ============================================================
__global__ void k_cg_init(const int* YS, float* x, float* r, float* p) {
    int t = blockIdx.x * blockDim.x + threadIdx.x;
    if (t >= NTOT * LDP) return;
    int i = t >> 5, c = t & 31;
    float v = (i < NS && c < WAYS && YS[i] == c) ? 1.f : 0.f;
    x[t] = 0.f; r[t] = v; p[t] = v;
}
__global__ void k_dot_rr(const float* r, float* rr) {
    int c = blockIdx.x;
    float s = 0.f;
    for (int i = threadIdx.x; i < NTOT; i += blockDim.x) { float v = r[i * LDP + c]; s += v * v; }
    s = bred(s);
    if (threadIdx.x == 0) rr[c] = s;
}

// V = S @ P  (f32 WMMA, M=4000, N=32, K=4000)
__global__ void k_spmv(const float* __restrict__ S, const float* __restrict__ P, float* __restrict__ V) {
    int lane = threadIdx.x & 31, wave = threadIdx.x >> 5;
    int row0 = blockIdx.x * 64 + (wave >> 1) * 16;
    if (row0 >= NTOT) return;                      // wave-uniform
    int col0 = (wave & 1) * 16;
    int m = lane & 15, n = lane & 15, kb = (lane >> 4) * 2;
    const float* arow = S + (size_t)(row0 + m) * NTOT;
    __builtin_prefetch(arow, 0, 1);
    v8f acc = {0.f, 0.f, 0.f, 0.f, 0.f, 0.f, 0.f, 0.f};
    #pragma unroll 4
    for (int k = 0; k < NTOT; k += 4) {
        v2f a = *reinterpret_cast<const v2f*>(arow + k + kb);
        v2f b;
        b[0] = P[(k + kb)     * LDP + col0 + n];
        b[1] = P[(k + kb + 1) * LDP + col0 + n];
        acc = __builtin_amdgcn_wmma_f32_16x16x4_f32(false, a, false, b, (short)0, acc, false, false);
    }
    int hi = (lane >> 4) * 8;
    #pragma unroll
    for (int v = 0; v < 8; ++v)
        V[(row0 + v + hi) * LDP + col0 + n] = acc[v];
}

__global__ void k_colsum(const float* p, float* cgs) {   // cs at cgs+128
    int c = blockIdx.x;
    float s = 0.f;
    for (int i = threadIdx.x; i < NTOT; i += blockDim.x) s += p[i * LDP + c];
    s = bred(s);
    if (threadIdx.x == 0) cgs[128 + c] = s;
}
__global__ void k_ap_pap(const float* p, const float* v, float* ap, float* cgs) {
    int c = blockIdx.x;
    float ej = EPS_C * cgs[128 + c];
    float s = 0.f;
    for (int i = threadIdx.x; i < NTOT; i += blockDim.x) {
        float pv = p[i * LDP + c];
        float av = pv - ALPHA_C * v[i * LDP + c] + ej;
        ap[i * LDP + c] = av;
        s += pv * av;
    }
    s = bred(s);
    if (threadIdx.x == 0) {
        cgs[32 + c] = s;
        cgs[64 + c] = (c < WAYS && s != 0.f) ? cgs[c] / s : 0.f;   // alpha
    }
}
__global__ void k_upd_xr(float* x, float* r, const float* p, const float* ap, float* cgs) {
    int c = blockIdx.x;
    float al = cgs[64 + c];
    float s = 0.f;
    for (int i = threadIdx.x; i < NTOT; i += blockDim.x) {
        int t = i * LDP + c;
        x[t] += al * p[t];
        float rn = r[t] - al * ap[t];
        r[t] = rn; s += rn * rn;
    }
    s = bred(s);
    if (threadIdx.x == 0) {
        float old = cgs[c];
        cgs[96 + c] = (c < WAYS && old != 0.f) ? s / old : 0.f;    // beta
        cgs[c] = s;
    }
}
__global__ void k_upd_p(float* p, const float* r, const float* cgs) {
    int c = blockIdx.x;
    float be = cgs[96 + c];
    for (int i = threadIdx.x; i < NTOT; i += blockDim.x) {
        int t = i * LDP + c;
        p[t] = r[t] + be * p[t];
    }
}

// ---------------- Fq top-K per class, center, proto ----------------
__global__ void k_topK_class(const float* x, int* qidx) {
    __shared__ float vals[NQ];
    __shared__ float sv[256];
    __shared__ int   si[256];
    int c = blockIdx.x, t = threadIdx.x;
    for (int q = t; q < NQ; q += blockDim.x) vals[q] = x[(NS + q) * LDP + c];
    __syncthreads();
    for (int g = 0; g < GK; ++g) {
        float bv = -1e30f; int bi = 0;
        for (int q = t; q < NQ; q += blockDim.x) {
            float v = vals[q];
            if (v > bv || (v == bv && q < bi)) { bv = v; bi = q; }
        }
        sv[t] = bv; si[t] = bi; __syncthreads();
        for (int s = blockDim.x >> 1; s > 0; s >>= 1) {
            if (t < s) {
                if (sv[t + s] > sv[t] || (sv[t + s] == sv[t] && si[t + s] < si[t])) {
                    sv[t] = sv[t + s]; si[t] = si[t + s];
                }
            }
            __syncthreads();
        }
        if (t == 0) { qidx[c * GK + g] = si[0]; vals[si[0]] = -1e30f; }
        __syncthreads();
    }
}
__global__ void k_center(const float* XQ, const int* qidx, float* center) {
    int t = blockIdx.x * blockDim.x + threadIdx.x;
    if (t >= WAYS * DIMD) return;
    int c = t / DIMD, d = t % DIMD;
    float s = 0.f;
    for (int g = 0; g < GK; ++g) s += XQ[qidx[c * GK + g] * DIMD + d];
    center[t] = s * (1.f / GK);
}
__global__ void k_proto(const float* XS, float* proto) {
    int t = blockIdx.x * blockDim.x + threadIdx.x;
    if (t >= WAYS * DIMD) return;
    int w = t / DIMD, d = t % DIMD;
    float s = 0.f;
    for (int sh = 0; sh < SHOTS; ++sh) s += XS[(w * SHOTS + sh) * DIMD + d];
    proto[t] = s * (1.f / SHOTS);
}

// ---------------- attention convs (per-d independent 1-D convs) ----------------
__global__ void k_as_conv1(const float* XS, const float* w1, const float* b1, float* h1a) {
    int t = blockIdx.x * blockDim.x + threadIdx.x;
    if (t >= WAYS * 32 * SHOTS * DIMD) return;
    int d = t % DIMD, r = t / DIMD;
    int h = r % SHOTS; r /= SHOTS;
    int c = r % 32, w = r / 32;
    float acc = b1[c];
    for (int kh = 0; kh < SHOTS; ++kh) {
        int hs = h + kh - 12;
        if (hs >= 0 && hs < SHOTS) acc += XS[(w * SHOTS + hs) * DIMD + d] * w1[c * SHOTS + kh];
    }
    h1a[t] = fmaxf(acc, 0.f);
}
__global__ void k_as_conv2(const float* h1a, const float* w2, const float* b2, float* h2a) {
    int t = blockIdx.x * blockDim.x + threadIdx.x;
    if (t >= WAYS * 64 * SHOTS * DIMD) return;
    int d = t % DIMD, r = t / DIMD;
    int h = r % SHOTS; r /= SHOTS;
    int c2 = r % 64, w = r / 64;
    float acc = b2[c2];
    for (int cin = 0; cin < 32; ++cin) {
        const float* src = h1a + ((size_t)(w * 32 + cin) * SHOTS) * DIMD + d;
        const float* wk = w2 + (c2 * 32 + cin) * SHOTS;
        for (int kh = 0; kh < SHOTS; ++kh) {
            int hs = h + kh - 12;
            if (hs >= 0 && hs < SHOTS) acc += src[hs * DIMD] * wk[kh];
        }
    }
    h2a[t] = fmaxf(acc, 0.f);
}
__global__ void k_as_conv3(const float* h2a, const float* wf, const float* bf, float* score) {
    int t = blockIdx.x * blockDim.x + threadIdx.x;
    if (t >= WAYS * DIMD) return;
    int w = t / DIMD, d = t % DIMD;
    float acc = bf[0];
    for (int cin = 0; cin < 64; ++cin)
        for (int kh = 0; kh < SHOTS; ++kh)
            acc += h2a[((size_t)(w * 64 + cin) * SHOTS + kh) * DIMD + d] * wf[cin * SHOTS + kh];
    score[t] = fmaxf(acc, 0.f);
}

// ---------------- auto_lambda ----------------
__global__ void k_al1(const float* proto, const float* center, const float* W1, const float* B1, float* hal) {
    int t = blockIdx.x * blockDim.x + threadIdx.x;
    if (t >= WAYS * 1200) return;
    int w = t / 1200, j = t % 1200;
    float s = B1[j];
    const float* pr = proto + w * DIMD;
    const float* ce = center + w * DIMD;
    const float* wr = W1 + (size_t)j * 600;
    for (int k = 0; k < DIMD; ++k) s += pr[k] * wr[k];
    for (int k = 0; k < DIMD; ++k) s += ce[k] * wr[DIMD + k];
    hal[t] = fmaxf(s, 0.f);
}
__global__ void k_al2(const float* hal, const float* W2, const float* B2, float* lam) {
    int w = blockIdx.x;
    float s = 0.f;
    for (int j = threadIdx.x; j < 1200; j += blockDim.x) s += hal[w * 1200 + j] * W2[j];
    s = bred(s);
    if (threadIdx.x == 0) lam[w] = 1.f / (1.f + expf(-(s + B2[0])));
}
__global__ void k_aup(const float* proto, const float* center, const float* lam, float* aup) {
    int t = blockIdx.x * blockDim.x + threadIdx.x;
    if (t >= WAYS * DIMD) return;
    int w = t / DIMD;
    float l = lam[w];
    aup[t] = (1.f - l) * proto[t] + l * center[t];
}

// ============================================================
// out3[q,w,c] = sum_d (XQ[q,d]-aup[w,d])^2 * score[c,d]  (f32 WMMA)
// ============================================================
__global__ void k_dist_wmma(const float* __restrict__ XQ, const float* __restrict__ aup,
                            const float* __restrict__ score, float* __restrict__ out3) {
    __shared__ float pw[DIMD];
    int w = blockIdx.y;
    for (int d = threadIdx.x; d < DIMD; d += blockDim.x) pw[d] = aup[w * DIMD + d];
    __syncthreads();
    int lane = threadIdx.x & 31, wave = threadIdx.x >> 5;
    int q0 = blockIdx.x * 128 + wave * 16;
    int m = lane & 15, n = lane & 15, kb = (lane >> 4) * 2;
    int q = q0 + m; if (q >= NQ) q = NQ - 1;     // clamp loads; stores guarded
    const float* xrow = XQ + (size_t)q * DIMD;
    v8f acc0 = {0.f, 0.f, 0.f, 0.f, 0.f, 0.f, 0.f, 0.f};
    v8f acc1 = {0.f, 0.f, 0.f, 0.f, 0.f, 0.f, 0.f, 0.f};
    bool hi_valid = (n + 16) < WAYS;
    #pragma unroll 5
    for (int k = 0; k < DIMD; k += 4) {
        int d0 = k + kb;
        v2f xq = *reinterpret_cast<const v2f*>(xrow + d0);
        v2f a;
        float t0 = xq[0] - pw[d0], t1 = xq[1] - pw[d0 + 1];
        a[0] = t0 * t0; a[1] = t1 * t1;
        v2f b0, b1;
        b0[0] = score[n * DIMD + d0];
        b0[1] = score[n * DIMD + d0 + 1];
        b1[0] = hi_valid ? score[(n + 16) * DIMD + d0] : 0.f;
        b1[1] = hi_valid ? score[(n + 16) * DIMD + d0 + 1] : 0.f;
        acc0 = __builtin_amdgcn_wmma_f32_16x16x4_f32(false, a, false, b0, (short)0, acc0, false, false);
        acc1 = __builtin_amdgcn_wmma_f32_16x16x4_f32(false, a, false, b1, (short)0, acc1, false, false);
    }
    int hi = (lane >> 4) * 8;
    #pragma unroll
    for (int v = 0; v < 8; ++v) {
        int qe = q0 + v + hi;
        if (qe < NQ) {
            out3[((size_t)qe * WAYS + w) * WAYS + n] = acc0[v];
            if (hi_valid) out3[((size_t)qe * WAYS + w) * WAYS + n + 16] = acc1[v];
        }
    }
}

__global__ void k_pred(const float* out3, float* pred) {
    int t = blockIdx.x * blockDim.x + threadIdx.x;
    if (t >= NQ * WAYS) return;
    float s = 0.f;
    const float* base = out3 + (size_t)t * WAYS;
    for (int c = 0; c < WAYS; ++c) s += sqrtf(fmaxf(base[c], 0.f));
    pred[t] = -s;
}

// ---------------- final losses / accuracy ----------------
__global__ void k_final(const float* x, const float* pred, const int* YS, const int* YQ, float* out) {
    int t = threadIdx.x;
    float s_lp = 0.f;
    for (int i = t; i < NTOT; i += blockDim.x) {
        const float* row = x + i * LDP;
        float mx = row[0];
        for (int c = 1; c < WAYS; ++c) mx = fmaxf(mx, row[c]);
        float se = 0.f;
        for (int c = 0; c < WAYS; ++c) se += expf(row[c] - mx);
        float lse = mx + logf(se);
        int g = (i < NS) ? YS[i] : YQ[i - NS];
        s_lp += lse - row[g];
    }
    s_lp = bred(s_lp);

    float s_q = 0.f, s_acc = 0.f;
    for (int q = t; q < NQ; q += blockDim.x) {
        const float* row = pred + q * WAYS;
        float mx = row[0]; int am = 0;
        for (int c = 1; c < WAYS; ++c) if (row[c] > mx) { mx = row[c]; am = c; }
        float se = 0.f;
        for (int c = 0; c < WAYS; ++c) se += expf(row[c] - mx);
        float lse = mx + logf(se);
        int y = YQ[q];
        s_q += lse - row[y];
        s_acc += (am == y) ? 1.f : 0.f;
    }
    s_q = bred(s_q);
    s_acc = bred(s_acc);
    if (t == 0) {
        out[0] = s_acc / NQ;
        out[1] = s_q / NQ + s_lp / NTOT;
    }
}

// ============================================================
extern "C" void kernel_launch(void* const* d_in, const int* in_sizes, int n_in,
                              void* d_out, int out_size, void* d_ws, size_t ws_size,
                              hipStream_t stream) {
    (void)in_sizes; (void)n_in; (void)out_size; (void)ws_size;
    const float* XS   = (const float*)d_in[0];
    const float* XQ   = (const float*)d_in[1];
    const float* c1w  = (const float*)d_in[2];
    const float* c1b  = (const float*)d_in[3];
    const float* bn1g = (const float*)d_in[4];
    const float* bn1b = (const float*)d_in[5];
    const float* c2w  = (const float*)d_in[6];
    const float* c2b  = (const float*)d_in[7];
    const float* bn2g = (const float*)d_in[8];
    const float* bn2b = (const float*)d_in[9];
    const float* f3w  = (const float*)d_in[10];
    const float* f3b  = (const float*)d_in[11];
    const float* f4w  = (const float*)d_in[12];
    const float* f4b  = (const float*)d_in[13];
    const float* asw1 = (const float*)d_in[14];
    const float* asb1 = (const float*)d_in[15];
    const float* asw2 = (const float*)d_in[16];
    const float* asb2 = (const float*)d_in[17];
    const float* aswf = (const float*)d_in[18];
    const float* asbf = (const float*)d_in[19];
    const float* alw1 = (const float*)d_in[20];
    const float* alb1 = (const float*)d_in[21];
    const float* alw2 = (const float*)d_in[22];
    const float* alb2 = (const float*)d_in[23];
    const int*   YS   = (const int*)d_in[24];
    const int*   YQ   = (const int*)d_in[25];
    float* out = (float*)d_out;

    char* ws = (char*)d_ws;
    float* MAT   = (float*)(ws + OFF_MAT);
    unsigned char* MASK = (unsigned char*)(ws + OFF_MASK);
    float* OUT3  = (float*)(ws + OFF_MASK);          // reuse after mask dead
    float* EB    = (float*)(ws + OFF_E);
    float* H1    = (float*)(ws + OFF_H1);
    float* H2    = (float*)(ws + OFF_H2);
    float* SIG   = (float*)(ws + OFF_SIG);
    float* N2    = (float*)(ws + OFF_N2);
    float* BN    = (float*)(ws + OFF_BN);
    int*   TOPG  = (int*)(ws + OFF_TOPG);
    float* DSI   = (float*)(ws + OFF_DSI);
    float* Rv    = (float*)(ws + OFF_R);
    float* Pv    = (float*)(ws + OFF_P);
    float* Vv    = (float*)(ws + OFF_V);
    float* APv   = (float*)(ws + OFF_AP);
    float* Xv    = (float*)(ws + OFF_X);
    float* CGS   = (float*)(ws + OFF_CGS);
    int*   QIDX  = (int*)(ws + OFF_QIDX);
    float* CEN   = (float*)(ws + OFF_CENTER);
    float* PRO   = (float*)(ws + OFF_PROTO);
    float* HAL   = (float*)(ws + OFF_HAL);
    float* LAM   = (float*)(ws + OFF_LAM);
    float* AUP   = (float*)(ws + OFF_AUP);
    float* SCORE = (float*)(ws + OFF_SCORE);
    float* PRED  = (float*)(ws + OFF_PRED);
    float* H1A   = MAT;                      // after CG (S dead)
    float* H2A   = (float*)(ws + OFF_MAT + 19200000);

    // --- graph_weight -> sigma ---
    k_gw_conv1<<<(NTOT * 75 * 4 + 255) / 256, 256, 0, stream>>>(XS, XQ, c1w, c1b, H1);
    k_gw_bn1<<<75, 256, 0, stream>>>(H1, bn1g, bn1b, BN);
    k_gw_conv2<<<(NTOT * 4 + 255) / 256, 256, 0, stream>>>(H1, BN, c2w, c2b, H2);
    k_gw_bn2<<<1, 256, 0, stream>>>(H2, bn2g, bn2b, BN);
    k_gw_head<<<(NTOT + 255) / 256, 256, 0, stream>>>(H2, BN, f3w, f3b, f4w, f4b, SIG);

    // --- E, n2, Gram->W (WMMA) ---
    k_E<<<NTOT, 128, 0, stream>>>(XS, XQ, SIG, EB, N2);
    k_gram<<<dim3(125, 125), 128, 0, stream>>>(EB, N2, MAT);

    // --- top-G mask, normalize -> S (in place) ---
    k_topG<<<NTOT, 256, 0, stream>>>(MAT, TOPG);
    k_mask_clear<<<(NTOT * NTOT / 4 + 255) / 256, 256, 0, stream>>>((unsigned int*)MASK);
    k_mask_scatter<<<(NTOT * GK + 255) / 256, 256, 0, stream>>>(TOPG, MASK);
    k_wmask<<<(int)(((size_t)NTOT * NTOT + 255) / 256), 256, 0, stream>>>(MAT, MASK);
    k_dv<<<NTOT, 256, 0, stream>>>(MAT, DSI);
    k_scaleS<<<(int)(((size_t)NTOT * NTOT + 255) / 256), 256, 0, stream>>>(MAT, DSI);

    // --- block CG: (I - alpha*S + eps*J) X = Y ---
    k_cg_init<<<(NTOT * LDP + 255) / 256, 256, 0, stream>>>(YS, Xv, Rv, Pv);
    k_dot_rr<<<LDP, 256, 0, stream>>>(Rv, CGS);
    for (int it = 0; it < CG_ITERS; ++it) {
        k_spmv<<<(NTOT + 63) / 64, 256, 0, stream>>>(MAT, Pv, Vv);
        k_colsum<<<LDP, 256, 0, stream>>>(Pv, CGS);
        k_ap_pap<<<LDP, 256, 0, stream>>>(Pv, Vv, APv, CGS);
        k_upd_xr<<<LDP, 256, 0, stream>>>(Xv, Rv, Pv, APv, CGS);
        k_upd_p<<<LDP, 256, 0, stream>>>(Pv, Rv, CGS);
    }

    // --- Fq top-K, center, proto ---
    k_topK_class<<<WAYS, 256, 0, stream>>>(Xv, QIDX);
    k_center<<<(WAYS * DIMD + 255) / 256, 256, 0, stream>>>(XQ, QIDX, CEN);
    k_proto<<<(WAYS * DIMD + 255) / 256, 256, 0, stream>>>(XS, PRO);

    // --- attention scores (S dead: reuse MAT region) ---
    k_as_conv1<<<(WAYS * 32 * SHOTS * DIMD + 255) / 256, 256, 0, stream>>>(XS, asw1, asb1, H1A);
    k_as_conv2<<<(WAYS * 64 * SHOTS * DIMD + 255) / 256, 256, 0, stream>>>(H1A, asw2, asb2, H2A);
    k_as_conv3<<<(WAYS * DIMD + 255) / 256, 256, 0, stream>>>(H2A, aswf, asbf, SCORE);

    // --- auto lambda, au_proto ---
    k_al1<<<(WAYS * 1200 + 255) / 256, 256, 0, stream>>>(PRO, CEN, alw1, alb1, HAL);
    k_al2<<<WAYS, 256, 0, stream>>>(HAL, alw2, alb2, LAM);
    k_aup<<<(WAYS * DIMD + 255) / 256, 256, 0, stream>>>(PRO, CEN, LAM, AUP);

    // --- weighted distances (WMMA) + pred ---
    k_dist_wmma<<<dim3((NQ + 127) / 128, WAYS), 256, 0, stream>>>(XQ, AUP, SCORE, OUT3);
    k_pred<<<(NQ * WAYS + 255) / 256, 256, 0, stream>>>(OUT3, PRED);

    // --- losses + accuracy ---
    k_final<<<1, 256, 0, stream>>>(Xv, PRED, YS, YQ, out);
}